// ReduceMask_70222715290212
// MI455X (gfx1250) — compile-verified
//
#include <hip/hip_runtime.h>
#include <stdint.h>

// Problem constants (reference: mask [16, 2048, 2048, 1] fp32, 16x16 max pool, TOL=0.5)
#define IMG_B   16
#define IMG_H   2048
#define IMG_W   2048
#define BC0     128               // H / 16
#define BC1     128               // W / 16
#define NBLOCKS (IMG_B * BC0 * BC1)   // 262144 pooled blocks
#define CHUNK   1024
#define NCHUNK  (NBLOCKS / CHUNK)     // 256

// ---------------------------------------------------------------------------
// Stage 1: max-pool 16x16 tiles via async global->LDS (CDNA5 ASYNCcnt path).
// One 256-thread WG handles a 16-row x 256-col tile (16 KB) = 16 blocks.
// ---------------------------------------------------------------------------
__global__ __launch_bounds__(256) void pool_kernel(const float* __restrict__ mask,
                                                   int* __restrict__ flags) {
    __shared__ __align__(16) float sh[16 * 256];   // tile, [row][col]
    __shared__ float part[256];                    // [row][block] partial maxima

    const int t   = threadIdx.x;
    const int w   = blockIdx.x;            // 16384 WGs = 16 * 128 * 8
    const int bi  = w / (BC0 * 8);         // image
    const int rem = w % (BC0 * 8);
    const int by  = rem / 8;               // block-row
    const int bxg = rem % 8;               // group of 16 blocks along W

    const float* tbase = mask + ((size_t)bi * IMG_H + (size_t)by * 16) * IMG_W
                              + (size_t)bxg * 256;

    // Low 32 bits of a generic pointer to LDS are the wave-relative LDS byte
    // address (flat aperture rule).
    const unsigned lbase = (unsigned)(uintptr_t)(&sh[0]);

    // 4 async b128 loads per thread: float4 index f = t + k*256 covers the
    // whole 16x256 tile, fully coalesced along W.
#pragma unroll
    for (int k = 0; k < 4; ++k) {
        const int f  = t + k * 256;
        const int r  = f >> 6;        // row 0..15
        const int c4 = f & 63;        // float4 within row
        const unsigned lds = lbase + (unsigned)(f * 16);
        const unsigned long long g =
            (unsigned long long)(uintptr_t)(tbase + (size_t)r * IMG_W + (size_t)c4 * 4);
        asm volatile("global_load_async_to_lds_b128 %0, %1, off"
                     :: "v"(lds), "v"(g) : "memory");
    }
    asm volatile("s_wait_asynccnt 0" ::: "memory");
    __syncthreads();

    // Reduce 1: per (row, block) max of 16 floats (4x ds_load_b128).
    const float4* sh4 = (const float4*)sh;
    const int r = t >> 4, b = t & 15;
    float4 a = sh4[r * 64 + b * 4 + 0];
    float4 c = sh4[r * 64 + b * 4 + 1];
    float4 d = sh4[r * 64 + b * 4 + 2];
    float4 e = sh4[r * 64 + b * 4 + 3];
    float m = fmaxf(fmaxf(fmaxf(a.x, a.y), fmaxf(a.z, a.w)),
                    fmaxf(fmaxf(c.x, c.y), fmaxf(c.z, c.w)));
    m = fmaxf(m, fmaxf(fmaxf(d.x, d.y), fmaxf(d.z, d.w)));
    m = fmaxf(m, fmaxf(fmaxf(e.x, e.y), fmaxf(e.z, e.w)));
    part[t] = m;                       // part[row*16 + block]
    __syncthreads();

    // Reduce 2: across 16 rows; write 0/1 flag per block.
    if (t < 16) {
        float mm = part[t];
#pragma unroll
        for (int rr = 1; rr < 16; ++rr) mm = fmaxf(mm, part[rr * 16 + t]);
        const int flat = bi * (BC0 * BC1) + by * BC1 + bxg * 16 + t;
        flags[flat] = (mm > 0.5f) ? 1 : 0;
    }
}

// ---------------------------------------------------------------------------
// Stage 2a: fill the indices region with -1 sentinels.
// ---------------------------------------------------------------------------
__global__ __launch_bounds__(256) void fill_kernel(int* __restrict__ out_idx) {
    const int i = blockIdx.x * 256 + threadIdx.x;   // 3072 * 256 = 786432
    out_idx[i] = -1;
}

// ---------------------------------------------------------------------------
// Stage 2b: per-chunk (1024 flags) active counts.
// ---------------------------------------------------------------------------
__global__ __launch_bounds__(256) void count_kernel(const int* __restrict__ flags,
                                                    int* __restrict__ counts) {
    __shared__ int s[256];
    const int c = blockIdx.x, t = threadIdx.x;
    const int4 v = ((const int4*)(flags + c * CHUNK))[t];
    s[t] = v.x + v.y + v.z + v.w;
    __syncthreads();
    for (int off = 128; off > 0; off >>= 1) {
        if (t < off) s[t] += s[t + off];
        __syncthreads();
    }
    if (t == 0) counts[c] = s[0];
}

// ---------------------------------------------------------------------------
// Stage 2c: exclusive scan of the 256 chunk counts; total -> d_out[0].
// ---------------------------------------------------------------------------
__global__ __launch_bounds__(256) void scan_kernel(const int* __restrict__ counts,
                                                   int* __restrict__ offs,
                                                   int* __restrict__ total_out) {
    __shared__ int s[256];
    const int t = threadIdx.x;
    const int v = counts[t];
    s[t] = v;
    __syncthreads();
    for (int off = 1; off < 256; off <<= 1) {
        const int x = (t >= off) ? s[t - off] : 0;
        __syncthreads();
        s[t] += x;
        __syncthreads();
    }
    offs[t] = s[t] - v;                // exclusive prefix
    if (t == 255) total_out[0] = s[255];
}

// ---------------------------------------------------------------------------
// Stage 2d: stable scatter of (b, y, x) rows in ascending flat order,
// matching jnp.nonzero semantics.
// ---------------------------------------------------------------------------
__global__ __launch_bounds__(256) void scatter_kernel(const int* __restrict__ flags,
                                                      const int* __restrict__ offs,
                                                      int* __restrict__ out_idx) {
    __shared__ int s[256];
    const int c = blockIdx.x, t = threadIdx.x;
    const int4 v = ((const int4*)(flags + c * CHUNK))[t];
    const int cnt = v.x + v.y + v.z + v.w;
    s[t] = cnt;
    __syncthreads();
    for (int off = 1; off < 256; off <<= 1) {
        const int x = (t >= off) ? s[t - off] : 0;
        __syncthreads();
        s[t] += x;
        __syncthreads();
    }
    int pos = offs[c] + (s[t] - cnt);  // exclusive within chunk
    const int g0 = c * CHUNK + t * 4;
    const int fl[4] = {v.x, v.y, v.z, v.w};
#pragma unroll
    for (int j = 0; j < 4; ++j) {
        if (fl[j]) {
            const int g = g0 + j;
            const int b = g >> 14;             // / (128*128)
            const int r = g & 16383;
            out_idx[pos * 3 + 0] = b;
            out_idx[pos * 3 + 1] = r >> 7;     // y
            out_idx[pos * 3 + 2] = r & 127;    // x
            ++pos;
        }
    }
}

extern "C" void kernel_launch(void* const* d_in, const int* in_sizes, int n_in,
                              void* d_out, int out_size, void* d_ws, size_t ws_size,
                              hipStream_t stream) {
    (void)in_sizes; (void)n_in; (void)out_size; (void)ws_size;
    const float* mask = (const float*)d_in[0];
    int* out = (int*)d_out;            // [0] = bin_counts, [1..] = indices (int32)

    int* flags  = (int*)d_ws;          // NBLOCKS ints (1 MiB)
    int* counts = flags + NBLOCKS;     // NCHUNK ints
    int* offs   = counts + NCHUNK;     // NCHUNK ints

    pool_kernel   <<<IMG_B * BC0 * 8, 256, 0, stream>>>(mask, flags);
    fill_kernel   <<<(NBLOCKS * 3) / 256, 256, 0, stream>>>(out + 1);
    count_kernel  <<<NCHUNK, 256, 0, stream>>>(flags, counts);
    scan_kernel   <<<1, 256, 0, stream>>>(counts, offs, out);
    scatter_kernel<<<NCHUNK, 256, 0, stream>>>(flags, offs, out + 1);
}